// biGatedGCNLayer2_67259187855861
// MI455X (gfx1250) — compile-verified
//
#include <hip/hip_runtime.h>
#include <math.h>

// Problem sizes (fixed by the reference)
#define NN 50000
#define NE 800000
#define N64 ((size_t)NN * 64)
#define E64 ((size_t)NE * 64)
#define BN_EPS 1e-5f

typedef __attribute__((ext_vector_type(2))) float v2f;
typedef __attribute__((ext_vector_type(8))) float v8f;

// ---------------------------------------------------------------------------
// V_WMMA_F32_16X16X4_F32 wrapper (fp32 matrix core, keeps reference precision)
// args: (neg_a, A, neg_b, B, c_mod, C, reuse_a, reuse_b)
// ---------------------------------------------------------------------------
__device__ __forceinline__ v8f wmma4(v2f a, v2f b, v8f c) {
  return __builtin_amdgcn_wmma_f32_16x16x4_f32(false, a, false, b, (short)0, c,
                                               false, false);
}

// Opaque zero: defeats LICM so per-tile LDS B loads stay inside the loop
// (otherwise the compiler hoists all 128 B values into registers and spills).
__device__ __forceinline__ int opaque_zero() {
  int z = 0;
  asm volatile("" : "+v"(z));
  return z;
}

// Cooperatively stage W[K x 64] into LDS, pair-interleaved so each B fragment
// {W[k][col], W[k+1][col]} is one contiguous 64-bit LDS load (conflict-free:
// lanes 0..15 access stride-8B addresses spanning 32 banks).
__device__ __forceinline__ void stage_w_lds(float* lw, const float* __restrict__ W,
                                            int kelems) {
  for (int idx = threadIdx.x; idx < kelems; idx += blockDim.x) {
    const int k = idx >> 6;
    const int col = idx & 63;
    lw[((k >> 1) * 64 + col) * 2 + (k & 1)] = W[idx];
  }
  __syncthreads();
}

// One 16-row tile: A fragments from global, B fragments from LDS (ds_load_b64).
// KS = K/4 k-steps; 4 n-tiles of 16 columns. `lw` must carry the opaque
// loop-variant offset so the DS loads are not hoisted.
template <int KS>
__device__ __forceinline__ void tile_mma_lds(const float* __restrict__ Arow,
                                             int hi, int m, const float* lw,
                                             v8f (&acc)[4]) {
  v2f a[KS];
#pragma unroll
  for (int s = 0; s < KS; ++s) {
    const int k0 = 4 * s + 2 * hi;
    a[s].x = Arow[k0];
    a[s].y = Arow[k0 + 1];
  }
#pragma unroll
  for (int s = 0; s < KS; ++s) {
    const int kp = 2 * s + hi;  // k0 >> 1
    const v2f* p = (const v2f*)lw + (size_t)kp * 64;
#pragma unroll
    for (int t = 0; t < 4; ++t) {
      const v2f b = p[t * 16 + m];
      acc[t] = wmma4(a[s], b, acc[t]);
    }
  }
}

// Epilogue: D row = r + 8*hi, col = t*16 + (lane&15); add bias, optional relu.
__device__ __forceinline__ void store_tile(float* __restrict__ out,
                                           const float* __restrict__ bias,
                                           int tile, int m, int hi,
                                           const v8f (&acc)[4], bool do_relu) {
#pragma unroll
  for (int r = 0; r < 8; ++r) {
    float* orow = out + (size_t)(tile * 16 + r + 8 * hi) * 64;
#pragma unroll
    for (int t = 0; t < 4; ++t) {
      const int col = t * 16 + m;
      float v = acc[t][r] + bias[col];
      if (do_relu) v = fmaxf(v, 0.0f);
      orow[col] = v;
    }
  }
}

// ---------------------------------------------------------------------------
// Kernel 1: five node GEMMs sharing A = h (Ah,Bh,Dh,Eh, S1=relu).
// Weight tile staged in LDS; persistent waves stream 16-row tiles.
// ---------------------------------------------------------------------------
__global__ void k_node_gemms(const float* __restrict__ h,
                             const float* WA, const float* bA, float* Ah,
                             const float* WB, const float* bB, float* Bh,
                             const float* WD, const float* bD, float* Dh,
                             const float* WE, const float* bE, float* Eh,
                             const float* Ws1, const float* bs1, float* S1) {
  __shared__ float lw[64 * 64];  // 16 KB
  const float* W; const float* bias; float* out; bool relu = false;
  switch (blockIdx.y) {
    case 0: W = WA;  bias = bA;  out = Ah; break;
    case 1: W = WB;  bias = bB;  out = Bh; break;
    case 2: W = WD;  bias = bD;  out = Dh; break;
    case 3: W = WE;  bias = bE;  out = Eh; break;
    default: W = Ws1; bias = bs1; out = S1; relu = true; break;
  }
  stage_w_lds(lw, W, 64 * 64);

  const int lane = threadIdx.x & 31;
  const int m = lane & 15;
  const int hi = lane >> 4;
  const int nwaves = gridDim.x * (blockDim.x >> 5);
  const int wid = blockIdx.x * (blockDim.x >> 5) + (threadIdx.x >> 5);

  for (int tile = wid; tile < NN / 16; tile += nwaves) {
    const float* lwv = lw + opaque_zero();  // keep DS loads in-loop
    const float* Arow = h + (size_t)(tile * 16 + m) * 64;
    v8f acc[4] = {v8f{}, v8f{}, v8f{}, v8f{}};
    tile_mma_lds<16>(Arow, hi, m, lwv, acc);
    store_tile(out, bias, tile, m, hi, acc, relu);
  }
}

// ---------------------------------------------------------------------------
// Kernel 2: Sh = softmax(S1 @ Ws2 + bs2)  -- one wave per row, warp softmax
// ---------------------------------------------------------------------------
__global__ void k_assign(const float* __restrict__ S1,
                         const float* __restrict__ Ws2,
                         const float* __restrict__ bs2,
                         float* __restrict__ Sh) {
  const int row = blockIdx.x * (blockDim.x >> 5) + (threadIdx.x >> 5);
  if (row >= NN) return;
  const int j = threadIdx.x & 31;
  const float* x = S1 + (size_t)row * 64;
  float acc = bs2[j];
#pragma unroll
  for (int k = 0; k < 64; ++k) acc += x[k] * Ws2[k * 32 + j];
  float mx = acc;
#pragma unroll
  for (int off = 16; off > 0; off >>= 1) mx = fmaxf(mx, __shfl_xor(mx, off));
  const float ex = __expf(acc - mx);
  float sm = ex;
#pragma unroll
  for (int off = 16; off > 0; off >>= 1) sm += __shfl_xor(sm, off);
  Sh[(size_t)row * 32 + j] = ex / sm;
}

// ---------------------------------------------------------------------------
// Kernel 3: Sq = Sh@WQ+bQ, Sk = Sh@WK+bK (K=32 WMMA, weights in LDS)
// ---------------------------------------------------------------------------
__global__ void k_qk(const float* __restrict__ Sh,
                     const float* WQ, const float* bQ, float* Sq,
                     const float* WK, const float* bK, float* Sk) {
  __shared__ float lw[32 * 64];  // 8 KB
  const float* W = (blockIdx.y == 0) ? WQ : WK;
  const float* bias = (blockIdx.y == 0) ? bQ : bK;
  float* out = (blockIdx.y == 0) ? Sq : Sk;
  stage_w_lds(lw, W, 32 * 64);

  const int lane = threadIdx.x & 31;
  const int m = lane & 15;
  const int hi = lane >> 4;
  const int nwaves = gridDim.x * (blockDim.x >> 5);
  const int wid = blockIdx.x * (blockDim.x >> 5) + (threadIdx.x >> 5);

  for (int tile = wid; tile < NN / 16; tile += nwaves) {
    const float* lwv = lw + opaque_zero();
    const float* Arow = Sh + (size_t)(tile * 16 + m) * 32;
    v8f acc[4] = {v8f{}, v8f{}, v8f{}, v8f{}};
    tile_mma_lds<8>(Arow, hi, m, lwv, acc);
    store_tile(out, bias, tile, m, hi, acc, false);
  }
}

// ---------------------------------------------------------------------------
// Kernel 4: att[e] = dot(Sk[src], Sq[dst]) / 64
// ---------------------------------------------------------------------------
__global__ void k_att(const int* __restrict__ src, const int* __restrict__ dst,
                      const float* __restrict__ Sk, const float* __restrict__ Sq,
                      float* __restrict__ att) {
  const int e = blockIdx.x * blockDim.x + threadIdx.x;
  if (e >= NE) return;
  const float4* a = (const float4*)(Sk + (size_t)src[e] * 64);
  const float4* b = (const float4*)(Sq + (size_t)dst[e] * 64);
  float acc = 0.0f;
#pragma unroll
  for (int i = 0; i < 16; ++i) {
    const float4 x = a[i], y = b[i];
    acc += x.x * y.x + x.y * y.y + x.z * y.z + x.w * y.w;
  }
  att[e] = acc * (1.0f / 64.0f);
}

// ---------------------------------------------------------------------------
// Kernel 5 (fused edge pipeline): Ce = e@WC+bC via WMMA (WC in LDS),
// then e_new = Dh[src]+Eh[dst]+Ce ; sigma = sigmoid(e_new)*att ; atomic
// segment sums into ssh/ss at dst. Accumulator row r lives on a 16-lane
// half-wave, so the src/dst gathers are fully coalesced (L2-resident).
// ---------------------------------------------------------------------------
__global__ void k_edge(const float* __restrict__ e,
                       const float* __restrict__ WC,
                       const float* __restrict__ bC,
                       const int* __restrict__ src, const int* __restrict__ dst,
                       const float* __restrict__ Dh, const float* __restrict__ Eh,
                       const float* __restrict__ Bh, const float* __restrict__ att,
                       float* __restrict__ e_new,
                       float* __restrict__ ssh, float* __restrict__ ss) {
  __shared__ float lw[64 * 64];  // 16 KB
  stage_w_lds(lw, WC, 64 * 64);

  const int lane = threadIdx.x & 31;
  const int m = lane & 15;
  const int hi = lane >> 4;
  const int nwaves = gridDim.x * (blockDim.x >> 5);
  const int wid = blockIdx.x * (blockDim.x >> 5) + (threadIdx.x >> 5);

  for (int tile = wid; tile < NE / 16; tile += nwaves) {
    const float* lwv = lw + opaque_zero();
    const float* Arow = e + (size_t)(tile * 16 + m) * 64;
    // Prefetch this wave's next tile of activations while WMMAs run.
    __builtin_prefetch(Arow + (size_t)nwaves * 16 * 64, 0, 0);
    v8f acc[4] = {v8f{}, v8f{}, v8f{}, v8f{}};
    tile_mma_lds<16>(Arow, hi, m, lwv, acc);
#pragma unroll
    for (int r = 0; r < 8; ++r) {
      const int edge = tile * 16 + r + 8 * hi;
      const int s = src[edge];
      const int d = dst[edge];
      const float at = att[edge];
      const float* Dhs = Dh + (size_t)s * 64;
      const float* Ehd = Eh + (size_t)d * 64;
      const float* Bhs = Bh + (size_t)s * 64;
      float* er = e_new + (size_t)edge * 64;
#pragma unroll
      for (int t = 0; t < 4; ++t) {
        const int col = t * 16 + m;
        const float v = acc[t][r] + bC[col] + Dhs[col] + Ehd[col];
        er[col] = v;
        const float sig = at / (1.0f + __expf(-v));  // sigmoid(v)*att
        atomicAdd(&ssh[(size_t)d * 64 + col], Bhs[col] * sig);
        atomicAdd(&ss[(size_t)d * 64 + col], sig);
      }
    }
  }
}

// ---------------------------------------------------------------------------
// Kernel 6: h_new = Ah + ssh/(ss + 1e-6)
// ---------------------------------------------------------------------------
__global__ void k_hnew(const float* __restrict__ Ah, const float* __restrict__ ssh,
                       const float* __restrict__ ss, float* __restrict__ h_new) {
  const size_t i = (size_t)blockIdx.x * blockDim.x + threadIdx.x;
  if (i >= N64) return;
  h_new[i] = Ah[i] + ssh[i] / (ss[i] + 1e-6f);
}

// ---------------------------------------------------------------------------
// Kernel 7: per-column sum / sumsq (block-reduced, then one atomic per col)
// ---------------------------------------------------------------------------
__global__ void k_stats(const float* __restrict__ X, int n_rows,
                        float* __restrict__ sum_out, float* __restrict__ sq_out) {
  __shared__ float s1[256];
  __shared__ float s2[256];
  const int c = threadIdx.x & 63;
  const int ry = threadIdx.x >> 6;  // 0..3
  float s = 0.0f, q = 0.0f;
  for (long long row = (long long)blockIdx.x * 4 + ry; row < n_rows;
       row += (long long)gridDim.x * 4) {
    const float v = X[row * 64 + c];
    s += v;
    q += v * v;
  }
  s1[threadIdx.x] = s;
  s2[threadIdx.x] = q;
  __syncthreads();
  if (threadIdx.x < 64) {
    s = s1[c] + s1[c + 64] + s1[c + 128] + s1[c + 192];
    q = s2[c] + s2[c + 64] + s2[c + 128] + s2[c + 192];
    atomicAdd(&sum_out[c], s);
    atomicAdd(&sq_out[c], q);
  }
}

// ---------------------------------------------------------------------------
// Kernel 8: finalize BN stats -> mu, rsqrt(var+eps) for h (tid<64) and e
// stats layout: [0]hsum [64]hsq [128]esum [192]esq [256]mu_h [320]rs_h
//               [384]mu_e [448]rs_e
// ---------------------------------------------------------------------------
__global__ void k_bnfinal(float* __restrict__ st) {
  const int tid = threadIdx.x;
  if (tid < 64) {
    const float inv_n = 1.0f / (float)NN;
    const float mu = st[tid] * inv_n;
    const float var = st[64 + tid] * inv_n - mu * mu;
    st[256 + tid] = mu;
    st[320 + tid] = rsqrtf(var + BN_EPS);
  } else if (tid < 128) {
    const int c = tid - 64;
    const float inv_n = 1.0f / (float)NE;
    const float mu = st[128 + c] * inv_n;
    const float var = st[192 + c] * inv_n - mu * mu;
    st[384 + c] = mu;
    st[448 + c] = rsqrtf(var + BN_EPS);
  }
}

// ---------------------------------------------------------------------------
// Kernel 9: out = residual + relu(bn(x))
// ---------------------------------------------------------------------------
__global__ void k_bn_relu_res(const float* __restrict__ x,
                              const float* __restrict__ res,
                              const float* __restrict__ mu,
                              const float* __restrict__ rs,
                              const float* __restrict__ gamma,
                              const float* __restrict__ beta,
                              float* __restrict__ out, size_t n_elem) {
  const size_t i = (size_t)blockIdx.x * blockDim.x + threadIdx.x;
  if (i >= n_elem) return;
  const int c = (int)(i & 63);
  const float v = (x[i] - mu[c]) * rs[c] * gamma[c] + beta[c];
  out[i] = res[i] + fmaxf(v, 0.0f);
}

// ---------------------------------------------------------------------------
extern "C" void kernel_launch(void* const* d_in, const int* in_sizes, int n_in,
                              void* d_out, int out_size, void* d_ws,
                              size_t ws_size, hipStream_t stream) {
  const float* h   = (const float*)d_in[0];
  const float* e   = (const float*)d_in[1];
  const int*   src = (const int*)d_in[2];
  const int*   dst = (const int*)d_in[3];
  const float* WA = (const float*)d_in[4];  const float* bA = (const float*)d_in[5];
  const float* WB = (const float*)d_in[6];  const float* bB = (const float*)d_in[7];
  const float* WC = (const float*)d_in[8];  const float* bC = (const float*)d_in[9];
  const float* WD = (const float*)d_in[10]; const float* bD = (const float*)d_in[11];
  const float* WE = (const float*)d_in[12]; const float* bE = (const float*)d_in[13];
  const float* Ws1 = (const float*)d_in[14]; const float* bs1 = (const float*)d_in[15];
  const float* Ws2 = (const float*)d_in[16]; const float* bs2 = (const float*)d_in[17];
  const float* WQ = (const float*)d_in[18]; const float* bQ = (const float*)d_in[19];
  const float* WK = (const float*)d_in[20]; const float* bK = (const float*)d_in[21];
  const float* gamma_h = (const float*)d_in[22]; const float* beta_h = (const float*)d_in[23];
  const float* gamma_e = (const float*)d_in[24]; const float* beta_e = (const float*)d_in[25];

  float* out = (float*)d_out;
  float* h_out = out;                 // [N,64]
  float* e_out = out + N64;           // [E,64]
  float* Sh    = out + N64 + E64;     // [N,32]

  // Workspace layout (floats); needs ~337 MB.
  float* ws = (float*)d_ws;
  float* Ah    = ws + 0 * N64;
  float* Bh    = ws + 1 * N64;
  float* Dh    = ws + 2 * N64;
  float* Eh    = ws + 3 * N64;
  float* S1    = ws + 4 * N64;
  float* Sq    = ws + 5 * N64;
  float* Sk    = ws + 6 * N64;
  float* ssh   = ws + 7 * N64;
  float* ss    = ws + 8 * N64;
  float* h_new = ws + 9 * N64;
  float* att   = ws + 10 * N64;
  float* e_new = ws + 10 * N64 + NE;
  float* stats = ws + 10 * N64 + NE + E64;  // 512 floats

  // Zero the atomic accumulation buffers (idempotent per call).
  hipMemsetAsync(ssh, 0, 2 * N64 * sizeof(float), stream);   // ssh + ss
  hipMemsetAsync(stats, 0, 512 * sizeof(float), stream);

  // 1) Five node GEMMs sharing A = h (fp32 WMMA, weights staged in LDS).
  dim3 g1(96, 5);  // 96 blocks x 8 waves = 768 persistent waves per weight
  k_node_gemms<<<g1, 256, 0, stream>>>(h, WA, bA, Ah, WB, bB, Bh,
                                       WD, bD, Dh, WE, bE, Eh, Ws1, bs1, S1);
  // 2) Sh = softmax(S1 @ Ws2 + bs2)  -> written straight into d_out.
  k_assign<<<(NN + 7) / 8, 256, 0, stream>>>(S1, Ws2, bs2, Sh);
  // 3) Sq / Sk (K=32 WMMA, weights staged in LDS).
  dim3 g3(96, 2);
  k_qk<<<g3, 256, 0, stream>>>(Sh, WQ, bQ, Sq, WK, bK, Sk);
  // 4) Edge attention scalar.
  k_att<<<(NE + 255) / 256, 256, 0, stream>>>(src, dst, Sk, Sq, att);
  // 5) Fused edge GEMM + gather + sigmoid*att + atomic segment sums.
  k_edge<<<1024, 256, 0, stream>>>(e, WC, bC, src, dst, Dh, Eh, Bh, att,
                                   e_new, ssh, ss);
  // 6) h_new.
  k_hnew<<<(int)((N64 + 255) / 256), 256, 0, stream>>>(Ah, ssh, ss, h_new);
  // 7) BN stats (column sums) for h_new and e_new.
  k_stats<<<1024, 256, 0, stream>>>(h_new, NN, stats + 0, stats + 64);
  k_stats<<<2048, 256, 0, stream>>>(e_new, NE, stats + 128, stats + 192);
  // 8) Finalize mu / inv_std.
  k_bnfinal<<<1, 128, 0, stream>>>(stats);
  // 9) Residual + relu(BN(.)) outputs.
  k_bn_relu_res<<<(int)((N64 + 255) / 256), 256, 0, stream>>>(
      h_new, h, stats + 256, stats + 320, gamma_h, beta_h, h_out, N64);
  k_bn_relu_res<<<(int)((E64 + 255) / 256), 256, 0, stream>>>(
      e_new, e, stats + 384, stats + 448, gamma_e, beta_e, e_out, E64);
}